// TransR_54846732370323
// MI455X (gfx1250) — compile-verified
//
#include <hip/hip_runtime.h>

typedef __attribute__((ext_vector_type(2))) float v2f;
typedef __attribute__((ext_vector_type(8))) float v8f;

#define ENT_SIZE   100
#define REL_SIZE   100
#define BATCH      2048
#define NEG        10
#define MARGIN     1.0f
#define WAVES_PB   8                      // 8 waves (wave32) = 256 threads/block
#define TRIPLES    (BATCH + BATCH * NEG)  // 22528, divisible by 8
#define KSTEPS     25                     // K = 100 = 25 * 4
#define RTILES     7                      // 7 * 16 = 112 >= 100 rel rows

// One wave per triple.  D = A x B with:
//   A (16x4) : row 0 = h slice, row 1 = t slice   (rows 2-15 = garbage, never read)
//   B (4x16) : column n = M[rt*16+n][k slice]     (cols >= 100 clamped, never read)
//   D (16x16): acc[0][n] = (M h)[rt*16+n], acc[1][n] = (M t)[rt*16+n]  for lanes n=0..15
__global__ __launch_bounds__(256)
void transr_score_kernel(const float* __restrict__ ent_emb,
                         const float* __restrict__ rel_emb,
                         const float* __restrict__ transfer,
                         const int*   __restrict__ pos_h,
                         const int*   __restrict__ pos_t,
                         const int*   __restrict__ pos_r,
                         const int*   __restrict__ neg_h,
                         const int*   __restrict__ neg_t,
                         const int*   __restrict__ neg_r,
                         float*       __restrict__ pos_score,
                         float*       __restrict__ neg_score)
{
    __shared__ float sht[WAVES_PB][2][ENT_SIZE];   // staged h (0) and t (1) per wave

    const int wave = threadIdx.x >> 5;
    const int lane = threadIdx.x & 31;
    const int trip = blockIdx.x * WAVES_PB + wave;  // grid is exact: trip < TRIPLES

    int hi, ti, ri;
    if (trip < BATCH) {
        hi = pos_h[trip]; ti = pos_t[trip]; ri = pos_r[trip];
    } else {
        const int j = trip - BATCH;
        hi = neg_h[j]; ti = neg_t[j]; ri = neg_r[j];
    }

    // Coalesced staging of the two gathered entity rows into LDS.
    const float* hrow = ent_emb + (long long)hi * ENT_SIZE;
    const float* trow = ent_emb + (long long)ti * ENT_SIZE;
    for (int e = lane; e < ENT_SIZE; e += 32) {
        sht[wave][0][e] = hrow[e];
        sht[wave][1][e] = trow[e];
    }
    __syncthreads();

    // A-operand lane mapping (ISA 7.12.2, 32-bit A 16x4):
    //   lanes 0-15 : row = lane&15, VGPRs hold K = {0,1}
    //   lanes 16-31: row = lane&15, VGPRs hold K = {2,3}
    const int koff   = (lane < 16) ? 0 : 2;
    const int rowSel = lane & 15;

    // Preload this lane's A values for ALL k-steps into registers, once.
    // Row 0 lanes need h, row 1 lanes need t; rows 2-15 read garbage (h or t by lane parity) — their
    // D rows are never consumed, so no masking is required.
    const float* ev = sht[wave][lane & 1];
    v2f areg[KSTEPS];
    #pragma unroll
    for (int ks = 0; ks < KSTEPS; ++ks) {
        areg[ks].x = ev[ks * 4 + koff + 0];
        areg[ks].y = ev[ks * 4 + koff + 1];
    }

    const float* M   = transfer + (long long)ri * (ENT_SIZE * REL_SIZE);
    const float* rel = rel_emb  + (long long)ri * REL_SIZE;

    float s = 0.0f;

    for (int rt = 0; rt < RTILES; ++rt) {
        const int row  = rt * 16 + rowSel;
        const int rowc = (row < REL_SIZE) ? row : (REL_SIZE - 1);   // address clamp only
        const float* Brow = M + rowc * ENT_SIZE + koff;             // 8-byte aligned

        // Warm L0 for the next row tile (global_prefetch_b8); M is L2-resident.
        {
            const int nrow  = rt * 16 + 16 + rowSel;
            const int nrowc = (nrow < REL_SIZE) ? nrow : (REL_SIZE - 1);
            __builtin_prefetch(M + nrowc * ENT_SIZE + koff, 0, 1);
        }

        // Two accumulators break the WMMA C-operand RAW chain.
        v8f acc0 = {}, acc1 = {};
        #pragma unroll
        for (int ks = 0; ks < KSTEPS; ++ks) {
            const v2f bv = *(const v2f*)(Brow + ks * 4);   // unconditional global_load_b64
            if (ks & 1)
                acc1 = __builtin_amdgcn_wmma_f32_16x16x4_f32(
                    false, areg[ks], false, bv, (short)0, acc1, false, false);
            else
                acc0 = __builtin_amdgcn_wmma_f32_16x16x4_f32(
                    false, areg[ks], false, bv, (short)0, acc0, false, false);
        }

        // D layout: lanes 0-15, VGPR v -> (m = v, n = lane).  m=0 -> Mh, m=1 -> Mt.
        const float ph = acc0[0] + acc1[0];
        const float pt = acc0[1] + acc1[1];
        if (lane < 16 && row < REL_SIZE)
            s += fabsf(ph + rel[row] - pt);
    }

    // Wave reduction (lanes 16-31 contribute 0).
    #pragma unroll
    for (int off = 16; off > 0; off >>= 1)
        s += __shfl_xor(s, off, 32);

    if (lane == 0) {
        if (trip < BATCH) pos_score[trip] = s;
        else              neg_score[trip - BATCH] = s;
    }
}

// Single-block hinge + batch reduction -> scalar loss.
__global__ __launch_bounds__(256)
void transr_loss_kernel(const float* __restrict__ pos_score,
                        const float* __restrict__ neg_score,
                        float* __restrict__ out)
{
    __shared__ float red[256];
    float local = 0.0f;
    for (int b = threadIdx.x; b < BATCH; b += 256) {
        const float p = pos_score[b];
        float n = 0.0f;
        #pragma unroll
        for (int k = 0; k < NEG; ++k) n += neg_score[b * NEG + k];
        n *= (1.0f / NEG);
        const float v = p - n + MARGIN;
        local += (v > 0.0f) ? v : 0.0f;
    }
    red[threadIdx.x] = local;
    __syncthreads();
    for (int off = 128; off > 0; off >>= 1) {
        if (threadIdx.x < off) red[threadIdx.x] += red[threadIdx.x + off];
        __syncthreads();
    }
    if (threadIdx.x == 0) out[0] = red[0];
}

extern "C" void kernel_launch(void* const* d_in, const int* in_sizes, int n_in,
                              void* d_out, int out_size, void* d_ws, size_t ws_size,
                              hipStream_t stream) {
    const float* ent_emb  = (const float*)d_in[0];
    const float* rel_emb  = (const float*)d_in[1];
    const float* transfer = (const float*)d_in[2];
    const int*   pos_h    = (const int*)d_in[3];
    const int*   pos_t    = (const int*)d_in[4];
    const int*   pos_r    = (const int*)d_in[5];
    const int*   neg_h    = (const int*)d_in[6];
    const int*   neg_t    = (const int*)d_in[7];
    const int*   neg_r    = (const int*)d_in[8];

    float* ws        = (float*)d_ws;
    float* pos_score = ws;             // BATCH floats
    float* neg_score = ws + BATCH;     // BATCH*NEG floats

    const int blocks = TRIPLES / WAVES_PB;   // 2816
    transr_score_kernel<<<blocks, 256, 0, stream>>>(
        ent_emb, rel_emb, transfer,
        pos_h, pos_t, pos_r, neg_h, neg_t, neg_r,
        pos_score, neg_score);

    transr_loss_kernel<<<1, 256, 0, stream>>>(pos_score, neg_score, (float*)d_out);
}